// Subtraction_68212670595965
// MI455X (gfx1250) — compile-verified
//
#include <hip/hip_runtime.h>
#include <cstdint>

// Subtraction op: x (16,64,56,56) f32 -> out (16,64,49,3136) f32
//   out[p, k, l] = x[p, oh, ow] - xpad[p, oh + k/7, ow + k%7],  pad=3
// Pure streaming: ~629 MB written, ~13 MB read. HBM-write bound (~28 us floor).

#define KK       7
#define PADV     3
#define HW       56
#define LPIX     (HW * HW)        // 3136 output pixels per (plane,k)
#define GROUPS   (LPIX / 4)       // 784 float4 groups per plane
#define LROWS    (HW + 2 * PADV)  // 62 padded rows
#define LSTRIDE  64               // padded row stride in LDS (floats), 16B-aligned cols
#define NTHREADS 256

typedef float vfloat4 __attribute__((ext_vector_type(4)));

// ---- CDNA5 async global->LDS path (guarded; fallback = load + ds_store) ----
#if defined(__has_builtin)
#if __has_builtin(__builtin_amdgcn_global_load_async_to_lds_b128) && \
    __has_builtin(__builtin_amdgcn_s_wait_asynccnt)
#define HAVE_ASYNC_LDS 1
#endif
#endif
#ifndef HAVE_ASYNC_LDS
#define HAVE_ASYNC_LDS 0
#endif

// Builtin wants GCC-style int vectors (vector_size), non-const, AS1/AS3.
typedef int v4i __attribute__((vector_size(16)));
typedef __attribute__((address_space(1))) v4i gv4i;
typedef __attribute__((address_space(3))) v4i lv4i;

__device__ __forceinline__ gv4i* to_glb_v4(const void* p) {
    return (gv4i*)(uintptr_t)p;
}
__device__ __forceinline__ lv4i* to_lds_v4(void* p) {
    // flat LDS address low 32 bits == LDS byte offset (aperture mapping)
    return (lv4i*)(uint32_t)(uintptr_t)p;
}

__global__ __launch_bounds__(NTHREADS) void
Subtraction_68212670595965_kernel(const float* __restrict__ x,
                                  float* __restrict__ out) {
    __shared__ __align__(16) float lds[LROWS * LSTRIDE];

    const int tid   = threadIdx.x;
    const int plane = blockIdx.x;                       // n*C + c, 0..1023
    const float* xin    = x   + (size_t)plane * (size_t)LPIX;
    float*       oplane = out + (size_t)plane * (size_t)(KK * KK * LPIX);

    // 1) Zero the whole padded tile (border stays 0 => no bounds checks later).
    {
        vfloat4 z4 = {0.f, 0.f, 0.f, 0.f};
        for (int z = tid; z < (LROWS * LSTRIDE) / 4; z += NTHREADS)
            *(vfloat4*)&lds[z * 4] = z4;
    }
    __syncthreads();

    // 2) Stage the 56x56 interior at (row+3, col+4): col+4 keeps B128 alignment.
    //    Interior is contiguous in global: group g covers floats [4g, 4g+4).
#if HAVE_ASYNC_LDS
    for (int g = tid; g < GROUPS; g += NTHREADS) {
        const int r  = g / 14;
        const int cq = (g % 14) * 4;
        __builtin_amdgcn_global_load_async_to_lds_b128(
            to_glb_v4(xin + 4 * g),
            to_lds_v4(&lds[(r + PADV) * LSTRIDE + 4 + cq]),
            0, 0);
    }
    __builtin_amdgcn_s_wait_asynccnt(0);
#else
    for (int g = tid; g < GROUPS; g += NTHREADS) {
        const int r  = g / 14;
        const int cq = (g % 14) * 4;
        vfloat4 v = *(const vfloat4*)(xin + 4 * g);
        *(vfloat4*)&lds[(r + PADV) * LSTRIDE + 4 + cq] = v;
    }
#endif
    __syncthreads();

    // 3) Per owned float4 pixel group: 7 rows x 3 aligned ds_load_b128 each
    //    (12-float window, cols owq..owq+11 cover patch cols owq+1..owq+10).
    //    Center = i==3 row, window slots [4..7] -> iterate i starting at 3.
    for (int g = tid; g < GROUPS; g += NTHREADS) {
        const int oh  = g / 14;          // 0..55
        const int owq = (g % 14) * 4;    // 0,4,...,52

        float* ob = oplane + 4 * g;

        float c0 = 0.f, c1 = 0.f, c2 = 0.f, c3 = 0.f;
        const int iorder[KK] = {3, 0, 1, 2, 4, 5, 6};

#pragma unroll
        for (int ii = 0; ii < KK; ++ii) {
            const int i = iorder[ii];
            // padded row oh+i-3 lives at LDS row oh+i; window starts at col owq
            const float* prow = &lds[(oh + i) * LSTRIDE + owq];
            const vfloat4 wa = *(const vfloat4*)(prow);
            const vfloat4 wb = *(const vfloat4*)(prow + 4);
            const vfloat4 wc = *(const vfloat4*)(prow + 8);
            const float w[12] = {wa.x, wa.y, wa.z, wa.w,
                                 wb.x, wb.y, wb.z, wb.w,
                                 wc.x, wc.y, wc.z, wc.w};
            if (ii == 0) {               // i == 3: center = x[oh][owq+0..3]
                c0 = w[4]; c1 = w[5]; c2 = w[6]; c3 = w[7];
            }

#pragma unroll
            for (int j = 0; j < KK; ++j) {
                // patch elem e at LDS col owq + e + j + 1 -> w[e + j + 1]
                vfloat4 v;
                v.x = c0 - w[j + 1];
                v.y = c1 - w[j + 2];
                v.z = c2 - w[j + 3];
                v.w = c3 - w[j + 4];
                __builtin_nontemporal_store(
                    v, (vfloat4*)(ob + (size_t)(i * KK + j) * (size_t)LPIX));
            }
        }
    }
}

extern "C" void kernel_launch(void* const* d_in, const int* in_sizes, int n_in,
                              void* d_out, int out_size, void* d_ws, size_t ws_size,
                              hipStream_t stream) {
    const float* x   = (const float*)d_in[0];
    float*       out = (float*)d_out;
    const int planes = in_sizes[0] / LPIX;   // 16*64 = 1024
    Subtraction_68212670595965_kernel<<<planes, NTHREADS, 0, stream>>>(x, out);
}